// GraphAttentionLayer_10307921510888
// MI455X (gfx1250) — compile-verified
//
#include <hip/hip_runtime.h>
#include <hip/hip_bf16.h>

typedef __attribute__((ext_vector_type(2))) float v2f;
typedef __attribute__((ext_vector_type(8))) float v8f;

#define N_NODES   8192
#define F_IN      512
#define F_OUT     256
#define ALPHA     0.2f
#define NROWS     16     // rows of output per attention workgroup
#define JCHUNK    128    // columns of attention processed per LDS stage
#define PSTRIDE   132    // padded row stride of P tile (floats): breaks bank conflicts

// ---------------------------------------------------------------------------
// Kernel 1: h = x @ W   (8192x512 @ 512x256), f32 WMMA 16x16x4.
// 8 waves / block, each wave computes one 16x16 output tile.
// ---------------------------------------------------------------------------
__launch_bounds__(256)
__global__ void gat_hW(const float* __restrict__ x,
                       const float* __restrict__ W,
                       float* __restrict__ h)
{
    const int lane  = threadIdx.x & 31;
    const int wave  = threadIdx.x >> 5;
    const int gid   = blockIdx.x * 8 + wave;      // 8192 tiles total
    const int m0    = (gid >> 4) << 4;            // 512 m-tiles
    const int n0    = (gid & 15) << 4;            // 16  n-tiles
    const int ml    = lane & 15;
    const int khalf = (lane >> 4) << 1;           // lanes 0-15 -> K0,K1 ; 16-31 -> K2,K3

    const float* xrow = x + (long)(m0 + ml) * F_IN;
    v8f c = {};
    for (int k = 0; k < F_IN; k += 4) {
        v2f a = *(const v2f*)(xrow + k + khalf);                 // A: 16x4 f32 fragment
        v2f b;
        b.x = W[(long)(k + khalf)     * F_OUT + n0 + ml];        // B: 4x16 f32 fragment
        b.y = W[(long)(k + khalf + 1) * F_OUT + n0 + ml];
        c = __builtin_amdgcn_wmma_f32_16x16x4_f32(false, a, false, b,
                                                  (short)0, c, false, false);
    }
    const int rofs = (lane >> 4) << 3;   // C layout: lanes 16-31 hold M = v+8
    #pragma unroll
    for (int v = 0; v < 8; ++v)
        h[(long)(m0 + v + rofs) * F_OUT + n0 + ml] = c[v];
}

// ---------------------------------------------------------------------------
// Kernel 2: s1 = h @ a[:256], s2 = h @ a[256:].  One wave32 per row.
// ---------------------------------------------------------------------------
__launch_bounds__(256)
__global__ void gat_scores(const float* __restrict__ h,
                           const float* __restrict__ a,
                           float* __restrict__ s1,
                           float* __restrict__ s2)
{
    const int lane = threadIdx.x & 31;
    const int wave = threadIdx.x >> 5;
    const int row  = blockIdx.x * 8 + wave;

    const float4* hv = (const float4*)(h + (long)row * F_OUT);
    const float4* a1 = (const float4*)(a);
    const float4* a2 = (const float4*)(a + F_OUT);

    float d1 = 0.f, d2 = 0.f;
    #pragma unroll
    for (int i = 0; i < 2; ++i) {                 // 256 cols / 32 lanes / 4 = 2 float4 each
        const int idx = lane * 2 + i;
        float4 hh = hv[idx];
        float4 b1 = a1[idx];
        float4 b2 = a2[idx];
        d1 += hh.x * b1.x + hh.y * b1.y + hh.z * b1.z + hh.w * b1.w;
        d2 += hh.x * b2.x + hh.y * b2.y + hh.z * b2.z + hh.w * b2.w;
    }
    #pragma unroll
    for (int off = 16; off > 0; off >>= 1) {
        d1 += __shfl_xor(d1, off);
        d2 += __shfl_xor(d2, off);
    }
    if (lane == 0) { s1[row] = d1; s2[row] = d2; }
}

// ---------------------------------------------------------------------------
// Async DMA of 16 bytes of adj into LDS (ASYNCcnt-tracked, no VGPR landing).
// VDST operand = LDS byte offset = low 32 bits of the generic shared pointer.
// ---------------------------------------------------------------------------
__device__ __forceinline__ void async_adj16(const int* gsrc, const int* ldst)
{
    unsigned lofs = (unsigned)(size_t)ldst;
    asm volatile("global_load_async_to_lds_b128 %0, %1, off"
                 :: "v"(lofs), "v"(gsrc) : "memory");
}

// ---------------------------------------------------------------------------
// Kernel 3: fused masked softmax + (attention @ h) + ELU.
// Block = 256 threads (8 waves), owns 16 output rows.
//   Pass A: stream adj row-block once, online (max, sum-exp) per row.
//   Pass B: double-buffered async adj staging in LDS; per 128-col chunk
//           build P tile in LDS, WMMA P(16x4) x h(4x16) while the next
//           adj chunk DMAs in behind the compute.
// Each wave owns a 32-column slice of the 256-wide output (2 accum tiles).
// ---------------------------------------------------------------------------
__launch_bounds__(256)
__global__ void gat_attn(const float* __restrict__ h,
                         const int*   __restrict__ adj,
                         const float* __restrict__ s1,
                         const float* __restrict__ s2,
                         float* __restrict__ out)
{
    __shared__ int   adjStage[2 * NROWS * JCHUNK];   // 2 x 8 KB double buffer
    __shared__ float P[NROWS * PSTRIDE];             // probability tile, padded stride
    __shared__ float redM[256], redS[256];
    __shared__ int   redC[256];
    __shared__ float rowM[NROWS], rowInvZ[NROWS], rowS1[NROWS];
    __shared__ int   rowFlag[NROWS];

    const int  t    = threadIdx.x;
    const int  lane = t & 31;
    const int  wave = t >> 5;
    const int  r    = t >> 4;       // local row 0..15 (16 threads per row)
    const int  c16  = t & 15;
    const long i0   = (long)blockIdx.x * NROWS;
    const long row  = i0 + r;

    const float s1r = s1[row];
    if (c16 == 0) rowS1[r] = s1r;

    // ---------------- Pass A: online masked (max, sum-exp) ----------------
    const int4* arow = (const int4*)(adj + row * (long)N_NODES);
    float m = -3.0e38f, s = 0.f;
    int cnt = 0;
    for (int j4 = c16; j4 < N_NODES / 4; j4 += 16) {
        __builtin_prefetch(arow + j4 + 64, 0, 0);    // global_prefetch_b8, 1KB ahead
        int4   av = arow[j4];
        float4 sv = *(const float4*)(s2 + j4 * 4);
        const int   aa[4] = { av.x, av.y, av.z, av.w };
        const float ss[4] = { sv.x, sv.y, sv.z, sv.w };
        #pragma unroll
        for (int q = 0; q < 4; ++q) {
            if (aa[q] > 0) {
                float e = s1r + ss[q];
                e = (e >= 0.f) ? e : ALPHA * e;          // leaky_relu
                ++cnt;
                if (e > m) { s = s * __expf(m - e) + 1.f; m = e; }
                else       { s += __expf(e - m); }
            }
        }
    }
    redM[t] = m; redS[t] = s; redC[t] = cnt;
    __syncthreads();

    if (t < NROWS) {   // combine 16 partials per row
        float mm = -3.0e38f, sacc = 0.f; int cc = 0;
        #pragma unroll
        for (int q = 0; q < 16; ++q) {
            const int idx = t * 16 + q;
            float mq = redM[idx], sq = redS[idx];
            cc += redC[idx];
            if (mq > mm) { sacc = sacc * __expf(mm - mq) + sq; mm = mq; }
            else         { sacc += sq * __expf(mq - mm); }
        }
        rowFlag[t] = (cc > 0);
        rowM[t]    = (cc > 0) ? mm : 0.f;
        // isolated node: softmax over uniform NEG values -> 1/N weights
        rowInvZ[t] = (cc > 0) ? 1.f / sacc : (1.f / (float)N_NODES);
    }
    __syncthreads();

    // ---------------- Pass B: async double-buffered chunks + WMMA ----------------
    v8f acc0 = {}, acc1 = {};
    const int n0    = wave * 32;            // this wave's 32-column output slice
    const int ml    = lane & 15;
    const int khalf = (lane >> 4) << 1;
    const int*  myAdj = adj + row * (long)N_NODES;
    const float mI  = rowM[r];
    const int   fl  = rowFlag[r];
    const float s1I = rowS1[r];
    const int   sofs = r * JCHUNK + c16 * 8;     // this thread's 8-int slice in a stage

    // preload chunk 0 into buffer 0
    async_adj16(myAdj + c16 * 8,     adjStage + sofs);
    async_adj16(myAdj + c16 * 8 + 4, adjStage + sofs + 4);

    int buf = 0;
    for (long j0 = 0; j0 < N_NODES; j0 += JCHUNK) {
        if (j0 + JCHUNK < N_NODES) {
            // kick off DMA of the next chunk into the other buffer
            const int* g = myAdj + j0 + JCHUNK + c16 * 8;
            int* l = adjStage + (buf ^ 1) * (NROWS * JCHUNK) + sofs;
            async_adj16(g,     l);
            async_adj16(g + 4, l + 4);
            // async loads complete in order: <=2 outstanding => current chunk landed
            asm volatile("s_wait_asynccnt 0x2" ::: "memory");
        } else {
            asm volatile("s_wait_asynccnt 0x0" ::: "memory");
        }
        __syncthreads();   // chunk j0 resident in adjStage[buf] for all waves

        // build P[r][0..127] for this chunk (each thread: 8 columns)
        {
            const int* stg = adjStage + buf * (NROWS * JCHUNK);
            #pragma unroll
            for (int piece = 0; piece < 2; ++piece) {
                const int j4 = c16 * 2 + piece;          // int4 index 0..31
                int4   av = *(const int4*)(stg + r * JCHUNK + j4 * 4);
                float4 sv = *(const float4*)(s2 + j0 + j4 * 4);
                const int   aa[4] = { av.x, av.y, av.z, av.w };
                const float sq[4] = { sv.x, sv.y, sv.z, sv.w };
                float4 pv;
                float pr[4];
                #pragma unroll
                for (int q = 0; q < 4; ++q) {
                    float e = s1I + sq[q];
                    e = (e >= 0.f) ? e : ALPHA * e;
                    float p = __expf(e - mI);
                    p = (aa[q] > 0) ? p : 0.f;
                    pr[q] = fl ? p : 1.f;                // isolated node -> uniform
                }
                pv.x = pr[0]; pv.y = pr[1]; pv.z = pr[2]; pv.w = pr[3];
                *(float4*)&P[r * PSTRIDE + j4 * 4] = pv;
            }
        }
        __syncthreads();   // P ready; also fences adjStage reads above vs next-iter DMA writes

        // P(16 x 128) @ h(128 x 32-slice) via f32 WMMA, K in steps of 4
        const float* hbase = h + j0 * F_OUT;
        #pragma unroll 8
        for (int kk = 0; kk < JCHUNK; kk += 4) {
            v2f a = *(const v2f*)&P[ml * PSTRIDE + kk + khalf];   // A frag from LDS
            const float* hb = hbase + (long)(kk + khalf) * F_OUT;
            v2f b0, b1;
            b0.x = hb[n0 + ml];         b0.y = hb[F_OUT + n0 + ml];
            b1.x = hb[n0 + 16 + ml];    b1.y = hb[F_OUT + n0 + 16 + ml];
            acc0 = __builtin_amdgcn_wmma_f32_16x16x4_f32(false, a, false, b0,
                                                         (short)0, acc0, false, false);
            acc1 = __builtin_amdgcn_wmma_f32_16x16x4_f32(false, a, false, b1,
                                                         (short)0, acc1, false, false);
        }
        buf ^= 1;
    }

    // ---------------- Epilogue: normalize + ELU ----------------
    const int rofs = (lane >> 4) << 3;
    #pragma unroll
    for (int v = 0; v < 8; ++v) {
        const int   rl = v + rofs;
        const float iz = rowInvZ[rl];
        float v0 = acc0[v] * iz;
        float v1 = acc1[v] * iz;
        v0 = (v0 > 0.f) ? v0 : expm1f(v0);
        v1 = (v1 > 0.f) ? v1 : expm1f(v1);
        float* orow = out + (i0 + rl) * F_OUT;
        orow[n0 + ml]      = v0;
        orow[n0 + 16 + ml] = v1;
    }
}

// ---------------------------------------------------------------------------
extern "C" void kernel_launch(void* const* d_in, const int* in_sizes, int n_in,
                              void* d_out, int out_size, void* d_ws, size_t ws_size,
                              hipStream_t stream)
{
    const float* x   = (const float*)d_in[0];   // (8192, 512)
    const int*   adj = (const int*)  d_in[1];   // (8192, 8192)
    const float* W   = (const float*)d_in[2];   // (512, 256)
    const float* a   = (const float*)d_in[3];   // (512, 1)
    float*       out = (float*)d_out;           // (8192, 256)

    float* h  = (float*)d_ws;                   // 8192*256 f32 = 8 MB
    float* s1 = h + (long)N_NODES * F_OUT;      // 8192 f32
    float* s2 = s1 + N_NODES;                   // 8192 f32

    gat_hW    <<<1024, 256, 0, stream>>>(x, W, h);
    gat_scores<<<1024, 256, 0, stream>>>(h, a, s1, s2);
    gat_attn  <<< 512, 256, 0, stream>>>(h, adj, s1, s2, out);
}